// NaiveLayerLSTM_83056077570356
// MI455X (gfx1250) — compile-verified
//
#include <hip/hip_runtime.h>
#include <hip/hip_bf16.h>
#include <cmath>

typedef __attribute__((ext_vector_type(2))) float v2f;
typedef __attribute__((ext_vector_type(8))) float v8f;

#define GBL __attribute__((address_space(1)))

#define HID 4096
#define KCHUNK 1024
#define NCHUNKS 8   // chunks 0..3 -> Wi with x, 4..7 -> Wh with h

// ---------------------------------------------------------------------------
// Phase 1: split-K GEMV via V_WMMA_F32_16X16X4_F32.
// grid = (HID/16 col tiles, 4 gates, 8 K-chunks), block = 1 wave (32 lanes).
// Each wave computes 16 output columns' partial dot product over 1024 K rows.
//
// A (16x4 f32) layout [ISA 7.12.2]: lane<16 holds K=0,1 ; lane>=16 holds K=2,3
//   -> broadcast vector segment: all 16 M-rows identical.
// B (4x16 f32) layout: lane = column n (mod 16), half-wave selects K rows
//   {0,1} / {2,3}, 2 VGPRs -> two coalesced 64B row segments per load.
// D rows are all identical; lane L<16 VGPR0 = D[0][L] = partial[col0+L].
// Dual accumulators break the serial WMMA RAW chain.
// ---------------------------------------------------------------------------
__global__ __launch_bounds__(32) void lstm_gates_wmma(
    const float* __restrict__ x, const float* __restrict__ h,
    const float* __restrict__ Wii, const float* __restrict__ Wif,
    const float* __restrict__ Wig, const float* __restrict__ Wio,
    const float* __restrict__ Whi, const float* __restrict__ Whf,
    const float* __restrict__ Whg, const float* __restrict__ Who,
    float* __restrict__ part)
{
    const int lane    = threadIdx.x;        // 0..31
    const int colTile = blockIdx.x;         // 0..255
    const int gate    = blockIdx.y;         // 0..3
    const int chunk   = blockIdx.z;         // 0..7

    // Uniform (SGPR) pointer selection -- no memory-indexed pointer arrays,
    // so no scratch and no flat loads.
    const float* W;
    const float* vec;
    if (chunk < 4) {
        vec = x;
        W = (gate == 0) ? Wii : (gate == 1) ? Wif : (gate == 2) ? Wig : Wio;
    } else {
        vec = h;
        W = (gate == 0) ? Whi : (gate == 1) ? Whf : (gate == 2) ? Whg : Who;
    }

    const int k0    = (chunk & 3) * KCHUNK;
    const int col0  = colTile * 16;
    const int n     = lane & 15;            // output column within tile
    const int khalf = (lane >> 4) << 1;     // 0 for lanes 0-15, 2 for 16-31

    // Force global address space for the streaming loads.
    const GBL float* wp = (const GBL float*)(W + (size_t)(k0 + khalf) * HID + col0 + n);
    const GBL float* vp = (const GBL float*)(vec + k0 + khalf);

    v8f acc0 = {}, acc1 = {};
    #pragma unroll 2
    for (int it = 0; it < KCHUNK / 8; ++it) {
        v2f a0, b0, a1, b1;
        a0.x = vp[0];            // broadcast vector values (cache hits)
        a0.y = vp[1];
        b0.x = wp[0];            // weight rows k, k+1 (64B coalesced / half-wave)
        b0.y = wp[HID];
        a1.x = vp[4];
        a1.y = vp[5];
        b1.x = wp[4 * HID];      // weight rows k+4, k+5
        b1.y = wp[5 * HID];
        acc0 = __builtin_amdgcn_wmma_f32_16x16x4_f32(
            false, a0, false, b0, (short)0, acc0, false, false);
        acc1 = __builtin_amdgcn_wmma_f32_16x16x4_f32(
            false, a1, false, b1, (short)0, acc1, false, false);
        vp += 8;
        wp += (size_t)8 * HID;
    }
    v8f acc = acc0 + acc1;

    // All D rows identical -> lane<16 VGPR0 holds partial for column col0+n.
    if (lane < 16) {
        GBL float* pp = (GBL float*)(part + (size_t)(chunk * 4 + gate) * HID + col0 + n);
        *pp = acc[0];
    }
}

// ---------------------------------------------------------------------------
// Phase 2: reduce 8 split-K partials per gate, add biases, gate math.
// ---------------------------------------------------------------------------
__global__ void lstm_pointwise(
    const float* __restrict__ part,
    const float* __restrict__ b_ii, const float* __restrict__ b_if,
    const float* __restrict__ b_ig, const float* __restrict__ b_io,
    const float* __restrict__ b_hi, const float* __restrict__ b_hf,
    const float* __restrict__ b_hg, const float* __restrict__ b_ho,
    const float* __restrict__ c_t, float* __restrict__ h_out)
{
    int j = blockIdx.x * blockDim.x + threadIdx.x;
    if (j >= HID) return;

    float g[4] = {0.f, 0.f, 0.f, 0.f};
    #pragma unroll
    for (int k = 0; k < NCHUNKS; ++k) {
        #pragma unroll
        for (int q = 0; q < 4; ++q)
            g[q] += part[(size_t)(k * 4 + q) * HID + j];
    }
    float gi = g[0] + b_ii[j] + b_hi[j];
    float gf = g[1] + b_if[j] + b_hf[j];
    float gg = g[2] + b_ig[j] + b_hg[j];
    float go = g[3] + b_io[j] + b_ho[j];

    float i_t = 1.f / (1.f + expf(-gi));
    float f_t = 1.f / (1.f + expf(-gf));
    float g_t = tanhf(gg);
    float o_t = 1.f / (1.f + expf(-go));

    float c_new = f_t * c_t[j] + i_t * g_t;
    h_out[j] = o_t * tanhf(c_new);
}

extern "C" void kernel_launch(void* const* d_in, const int* in_sizes, int n_in,
                              void* d_out, int out_size, void* d_ws, size_t ws_size,
                              hipStream_t stream) {
    const float* x    = (const float*)d_in[0];
    const float* Wii  = (const float*)d_in[1];
    const float* Wif  = (const float*)d_in[2];
    const float* Wig  = (const float*)d_in[3];
    const float* Wio  = (const float*)d_in[4];
    const float* Whi  = (const float*)d_in[5];
    const float* Whf  = (const float*)d_in[6];
    const float* Whg  = (const float*)d_in[7];
    const float* Who  = (const float*)d_in[8];
    const float* b_ii = (const float*)d_in[9];
    const float* b_if = (const float*)d_in[10];
    const float* b_ig = (const float*)d_in[11];
    const float* b_io = (const float*)d_in[12];
    const float* b_hi = (const float*)d_in[13];
    const float* b_hf = (const float*)d_in[14];
    const float* b_hg = (const float*)d_in[15];
    const float* b_ho = (const float*)d_in[16];
    const float* h_t  = (const float*)d_in[17];
    const float* c_t  = (const float*)d_in[18];

    float* part  = (float*)d_ws;            // 8 chunks * 4 gates * 4096 f32 = 512 KB
    float* h_out = (float*)d_out;

    dim3 grid1(HID / 16, 4, NCHUNKS);
    lstm_gates_wmma<<<grid1, 32, 0, stream>>>(
        x, h_t, Wii, Wif, Wig, Wio, Whi, Whf, Whg, Who, part);

    lstm_pointwise<<<HID / 256, 256, 0, stream>>>(
        part, b_ii, b_if, b_ig, b_io, b_hi, b_hf, b_hg, b_ho, c_t, h_out);
}